// Complete_process_54657753808968
// MI455X (gfx1250) — compile-verified
//
#include <hip/hip_runtime.h>
#include <stdint.h>

// Problem dims (from reference): L=16384, F=4096, H=64, C=128, K=100
#define L_DIM 16384
#define F_DIM 4096
#define H_DIM 64
#define C_DIM 128
#define K_SEL 100
#define M_PAD 112   // 7 WMMA M-tiles of 16

typedef __attribute__((ext_vector_type(2))) float v2f;
typedef __attribute__((ext_vector_type(8))) float v8f;

// ---------------------------------------------------------------------------
// Kernel 1: row sums over input_data [L,F]  (one wave per row, float4 loads)
// also initializes row_map to -1.
// ---------------------------------------------------------------------------
__global__ __launch_bounds__(256) void k_rowsum(const float* __restrict__ x,
                                                float* __restrict__ row_scores,
                                                int* __restrict__ row_map) {
  const int wave = threadIdx.x >> 5;
  const int lane = threadIdx.x & 31;
  const int row  = blockIdx.x * 8 + wave;
  const float4* p = (const float4*)(x + (size_t)row * F_DIM);
  float s = 0.f;
  for (int i = lane; i < F_DIM / 4; i += 32) {
    float4 v = p[i];
    s += v.x + v.y + v.z + v.w;
  }
  for (int off = 16; off; off >>= 1) s += __shfl_down(s, off, 32);
  if (lane == 0) { row_scores[row] = s; row_map[row] = -1; }
}

// ---------------------------------------------------------------------------
// Kernel 2: top-100 (lowest-index tie-break like jax.lax.top_k), single block,
// scores resident in LDS (64 KB of the 320 KB WGP LDS).
// Writes topk[0..99] (+pad to 112) and row_map[winner]=rank.
// ---------------------------------------------------------------------------
__global__ __launch_bounds__(1024) void k_topk(const float* __restrict__ scores,
                                               int* __restrict__ topk,
                                               int* __restrict__ row_map) {
  __shared__ float sc[L_DIM];
  __shared__ float rv[32];
  __shared__ int   ri[32];
  const int tid  = threadIdx.x;
  const int lane = tid & 31, wave = tid >> 5;
  for (int i = tid; i < L_DIM; i += 1024) sc[i] = scores[i];
  __syncthreads();
  for (int k = 0; k < K_SEL; ++k) {
    float bv = -INFINITY; int bi = 0x7fffffff;
    for (int i = tid; i < L_DIM; i += 1024) {
      float v = sc[i];
      if (v > bv) { bv = v; bi = i; }
    }
    for (int off = 16; off; off >>= 1) {
      float ov = __shfl_xor(bv, off, 32);
      int   oi = __shfl_xor(bi, off, 32);
      if (ov > bv || (ov == bv && oi < bi)) { bv = ov; bi = oi; }
    }
    if (lane == 0) { rv[wave] = bv; ri[wave] = bi; }
    __syncthreads();
    if (wave == 0) {
      bv = rv[lane]; bi = ri[lane];
      for (int off = 16; off; off >>= 1) {
        float ov = __shfl_xor(bv, off, 32);
        int   oi = __shfl_xor(bi, off, 32);
        if (ov > bv || (ov == bv && oi < bi)) { bv = ov; bi = oi; }
      }
      if (lane == 0) { topk[k] = bi; row_map[bi] = k; sc[bi] = -INFINITY; }
    }
    __syncthreads();
  }
  if (tid >= K_SEL && tid < M_PAD) topk[tid] = topk[0];  // pad for WMMA M-tiles
}

// ---------------------------------------------------------------------------
// Kernel 3a: code_emb = relu(code @ W_code + b_code)   [128]x[128,64] -> [64]
// ---------------------------------------------------------------------------
__global__ __launch_bounds__(64) void k_code_emb(const float* __restrict__ code,
                                                 const float* __restrict__ W_code,
                                                 const float* __restrict__ b_code,
                                                 float* __restrict__ code_emb) {
  const int h = threadIdx.x;
  float s = b_code[h];
  for (int c = 0; c < C_DIM; ++c) s += code[c] * W_code[c * H_DIM + h];
  code_emb[h] = fmaxf(s, 0.f);
}

// ---------------------------------------------------------------------------
// Kernel 3b: inp_emb = relu(rows @ W_in + b_in)  [112(pad),4096]x[4096,64]
// f32 WMMA 16x16x4: one wave per 16x16 tile; gridDim.x = 4 N-tiles,
// 7 waves = 7 M-tiles. A-frag gathers rows via topk[]; padded rows zeroed.
// ---------------------------------------------------------------------------
__global__ __launch_bounds__(224) void k_inp_emb(const float* __restrict__ x,
                                                 const int* __restrict__ topk,
                                                 const float* __restrict__ W_in,
                                                 const float* __restrict__ b_in,
                                                 float* __restrict__ inp_emb) {
  const int wave  = threadIdx.x >> 5;     // M-tile 0..6
  const int lane  = threadIdx.x & 31;
  const int tn    = blockIdx.x;           // N-tile 0..3
  const int mrow  = wave * 16 + (lane & 15);
  const int srow  = topk[mrow];
  const float keep = (mrow < K_SEL) ? 1.f : 0.f;
  const int hioff = (lane >> 4) ? 2 : 0;  // lanes 16-31 hold K=2,3
  const float* arow = x + (size_t)srow * F_DIM + hioff;
  const int ncol  = tn * 16 + (lane & 15);
  const float* bcol = W_in + (size_t)hioff * H_DIM + ncol;
  v8f acc = {};
  for (int k0 = 0; k0 < F_DIM; k0 += 4) {
    v2f a, b;
    a[0] = arow[k0]     * keep;
    a[1] = arow[k0 + 1] * keep;
    b[0] = bcol[(size_t)k0 * H_DIM];
    b[1] = bcol[(size_t)(k0 + 1) * H_DIM];
    acc = __builtin_amdgcn_wmma_f32_16x16x4_f32(false, a, false, b,
                                                (short)0, acc, false, false);
  }
  const float bias = b_in[ncol];
  const int mbase = wave * 16 + ((lane >> 4) ? 8 : 0);
  #pragma unroll
  for (int r = 0; r < 8; ++r)
    inp_emb[(mbase + r) * H_DIM + ncol] = fmaxf(acc[r] + bias, 0.f);
}

// ---------------------------------------------------------------------------
// Kernel 3c: enc_out = [inp_emb | code_emb] @ W_enc + b_enc
// [112,128]x[128,4096]: 256 blocks (N-tiles) x 7 waves (M-tiles), K in 4s.
// ---------------------------------------------------------------------------
__global__ __launch_bounds__(224) void k_enc(const float* __restrict__ inp_emb,
                                             const float* __restrict__ code_emb,
                                             const float* __restrict__ W_enc,
                                             const float* __restrict__ b_enc,
                                             float* __restrict__ enc_out) {
  const int wave  = threadIdx.x >> 5;
  const int lane  = threadIdx.x & 31;
  const int tn    = blockIdx.x;           // 0..255
  const int mrow  = wave * 16 + (lane & 15);
  const int hioff = (lane >> 4) ? 2 : 0;
  const int ncol  = tn * 16 + (lane & 15);
  v8f acc = {};
  for (int k0 = 0; k0 < 2 * H_DIM; k0 += 4) {
    const int ka = k0 + hioff;            // pairs never straddle the concat seam
    v2f a, b;
    a[0] = (ka < H_DIM) ? inp_emb[mrow * H_DIM + ka]     : code_emb[ka - H_DIM];
    a[1] = (ka < H_DIM) ? inp_emb[mrow * H_DIM + ka + 1] : code_emb[ka + 1 - H_DIM];
    b[0] = W_enc[(size_t)ka * F_DIM + ncol];
    b[1] = W_enc[(size_t)(ka + 1) * F_DIM + ncol];
    acc = __builtin_amdgcn_wmma_f32_16x16x4_f32(false, a, false, b,
                                                (short)0, acc, false, false);
  }
  const float bias = b_enc[ncol];
  const int mbase = wave * 16 + ((lane >> 4) ? 8 : 0);
  #pragma unroll
  for (int r = 0; r < 8; ++r)
    enc_out[(size_t)(mbase + r) * F_DIM + ncol] = acc[r] + bias;
}

// ---------------------------------------------------------------------------
// Kernel 4: Gumbel-max categorical per row, JAX threefry2x32 (key (0,42)).
// Flat element i pairs with i + 2^25, i.e. row r pairs with row r+8192:
// one wave handles the row pair, one threefry call per element pair.
// ---------------------------------------------------------------------------
__device__ __forceinline__ uint32_t rotl32(uint32_t x, int r) {
  return (x << r) | (x >> (32 - r));
}

__device__ __forceinline__ void threefry2x32(uint32_t x0, uint32_t x1,
                                             uint32_t& o0, uint32_t& o1) {
  const uint32_t k0 = 0u, k1 = 42u, k2 = 0x1BD11BDAu ^ k0 ^ k1;
  x0 += k0; x1 += k1;
#define TF_R(r) { x0 += x1; x1 = rotl32(x1, (r)); x1 ^= x0; }
  TF_R(13) TF_R(15) TF_R(26) TF_R(6)
  x0 += k1; x1 += k2 + 1u;
  TF_R(17) TF_R(29) TF_R(16) TF_R(24)
  x0 += k2; x1 += k0 + 2u;
  TF_R(13) TF_R(15) TF_R(26) TF_R(6)
  x0 += k0; x1 += k1 + 3u;
  TF_R(17) TF_R(29) TF_R(16) TF_R(24)
  x0 += k1; x1 += k2 + 4u;
  TF_R(13) TF_R(15) TF_R(26) TF_R(6)
  x0 += k2; x1 += k0 + 5u;
#undef TF_R
  o0 = x0; o1 = x1;
}

__device__ __forceinline__ float gumbel_from_bits(uint32_t bits) {
  // JAX uniform: bits>>9 | 1.0f  ->  [1,2) - 1  ->  scale to [tiny, 1)
  float f = __uint_as_float(0x3f800000u | (bits >> 9)) - 1.0f;
  const float tiny = 1.1754943508222875e-38f;
  float u = f * (1.0f - tiny) + tiny;
  return -__logf(-__logf(u));
}

__global__ __launch_bounds__(256) void k_sample(const float* __restrict__ x,
                                                const float* __restrict__ enc_out,
                                                const int* __restrict__ row_map,
                                                float* __restrict__ s_arr) {
  const int wave = threadIdx.x >> 5;
  const int lane = threadIdx.x & 31;
  const int r0 = blockIdx.x * 8 + wave;     // 0..8191
  const int r1 = r0 + L_DIM / 2;
  const int m0 = row_map[r0], m1 = row_map[r1];
  const float4* p0 = (const float4*)(m0 >= 0 ? enc_out + (size_t)m0 * F_DIM
                                             : x + (size_t)r0 * F_DIM);
  const float4* p1 = (const float4*)(m1 >= 0 ? enc_out + (size_t)m1 * F_DIM
                                             : x + (size_t)r1 * F_DIM);
  const uint32_t base = (uint32_t)r0 * (uint32_t)F_DIM;
  const uint32_t halfN = (uint32_t)(L_DIM / 2) * (uint32_t)F_DIM;  // 2^25
  float bv0 = -INFINITY, bv1 = -INFINITY;
  int bi0 = 0x7fffffff, bi1 = 0x7fffffff;
  for (int t = 0; t < (F_DIM / 4) / 32; ++t) {
    const int q = t * 32 + lane;            // float4 index: ascending per lane
    float4 a = p0[q];
    float4 b = p1[q];
    const float va[4] = {a.x, a.y, a.z, a.w};
    const float vb[4] = {b.x, b.y, b.z, b.w};
    #pragma unroll
    for (int e = 0; e < 4; ++e) {
      const int col = q * 4 + e;
      uint32_t o0, o1;
      threefry2x32(base + (uint32_t)col, base + (uint32_t)col + halfN, o0, o1);
      const float g0 = va[e] + gumbel_from_bits(o0);
      const float g1 = vb[e] + gumbel_from_bits(o1);
      if (g0 > bv0) { bv0 = g0; bi0 = col; }
      if (g1 > bv1) { bv1 = g1; bi1 = col; }
    }
  }
  for (int off = 16; off; off >>= 1) {
    float ov = __shfl_xor(bv0, off, 32); int oi = __shfl_xor(bi0, off, 32);
    if (ov > bv0 || (ov == bv0 && oi < bi0)) { bv0 = ov; bi0 = oi; }
    ov = __shfl_xor(bv1, off, 32); oi = __shfl_xor(bi1, off, 32);
    if (ov > bv1 || (ov == bv1 && oi < bi1)) { bv1 = ov; bi1 = oi; }
  }
  if (lane == 0) { s_arr[r0] = (float)bi0; s_arr[r1] = (float)bi1; }
}

// ---------------------------------------------------------------------------
// Kernel 5a/5b: out = s @ W_dec + b_dec, then L2-normalize. Deterministic
// two-stage reduction (128 blocks x 128 rows, then fixed-order combine).
// ---------------------------------------------------------------------------
__global__ __launch_bounds__(128) void k_dec_partial(const float* __restrict__ s_arr,
                                                     const float* __restrict__ W_dec,
                                                     float* __restrict__ partials) {
  const int c = threadIdx.x;
  const int b = blockIdx.x;                 // 0..127
  const int l0 = b * (L_DIM / 128);
  float acc = 0.f;
  for (int i = 0; i < L_DIM / 128; ++i) {
    const int l = l0 + i;
    acc += s_arr[l] * W_dec[(size_t)l * C_DIM + c];
  }
  partials[b * C_DIM + c] = acc;
}

__global__ __launch_bounds__(128) void k_finalize(const float* __restrict__ partials,
                                                  const float* __restrict__ b_dec,
                                                  float* __restrict__ out) {
  __shared__ float sh[128];
  const int c = threadIdx.x;
  float tot = b_dec[c];
  for (int b = 0; b < 128; ++b) tot += partials[b * C_DIM + c];
  sh[c] = tot * tot;
  __syncthreads();
  for (int off = 64; off; off >>= 1) {
    if (c < off) sh[c] += sh[c + off];
    __syncthreads();
  }
  const float denom = fmaxf(sqrtf(sh[0]), 1e-12f);
  out[c] = tot / denom;
}

// ---------------------------------------------------------------------------
extern "C" void kernel_launch(void* const* d_in, const int* in_sizes, int n_in,
                              void* d_out, int out_size, void* d_ws, size_t ws_size,
                              hipStream_t stream) {
  (void)in_sizes; (void)n_in; (void)out_size; (void)ws_size;
  const float* input_data = (const float*)d_in[0];
  const float* code   = (const float*)d_in[1];
  const float* W_in   = (const float*)d_in[2];
  const float* b_in   = (const float*)d_in[3];
  const float* W_code = (const float*)d_in[4];
  const float* b_code = (const float*)d_in[5];
  const float* W_enc  = (const float*)d_in[6];
  const float* b_enc  = (const float*)d_in[7];
  const float* W_dec  = (const float*)d_in[8];
  const float* b_dec  = (const float*)d_in[9];
  float* out = (float*)d_out;

  float* ws         = (float*)d_ws;              // ~2.03 MB total
  float* row_scores = ws;                        // 16384 f
  int*   row_map    = (int*)(ws + 16384);        // 16384 i
  int*   topk       = (int*)(ws + 32768);        // 128 i (100 + pad)
  float* code_emb   = ws + 32896;                // 64 f
  float* inp_emb    = ws + 32960;                // 112*64   = 7168 f
  float* enc_out    = ws + 40128;                // 112*4096 = 458752 f
  float* s_arr      = ws + 498880;               // 16384 f
  float* partials   = ws + 515264;               // 128*128  = 16384 f

  k_rowsum     <<<L_DIM / 8, 256, 0, stream>>>(input_data, row_scores, row_map);
  k_topk       <<<1, 1024, 0, stream>>>(row_scores, topk, row_map);
  k_code_emb   <<<1, 64, 0, stream>>>(code, W_code, b_code, code_emb);
  k_inp_emb    <<<4, 224, 0, stream>>>(input_data, topk, W_in, b_in, inp_emb);
  k_enc        <<<256, 224, 0, stream>>>(inp_emb, code_emb, W_enc, b_enc, enc_out);
  k_sample     <<<(L_DIM / 2) / 8, 256, 0, stream>>>(input_data, enc_out, row_map, s_arr);
  k_dec_partial<<<128, 128, 0, stream>>>(s_arr, W_dec, partials);
  k_finalize   <<<1, 128, 0, stream>>>(partials, b_dec, out);
}